// LinkPredictionModel_21242908246156
// MI455X (gfx1250) — compile-verified
//
#include <hip/hip_runtime.h>
#include <hip/hip_bf16.h>

typedef float v2f __attribute__((ext_vector_type(2)));
typedef float v8f __attribute__((ext_vector_type(8)));

#define N_NODES 20000
#define N_EDGES 320000
#define FEATS   256
#define LDA     (FEATS + 4)   // padded LDS row stride: 260 % 64 == 4 -> conflict-free

// ---------------------------------------------------------------------------
// Zero a float buffer
// ---------------------------------------------------------------------------
__global__ void zero_kernel(float* __restrict__ p, long n) {
    long i = (long)blockIdx.x * blockDim.x + threadIdx.x;
    if (i < n) p[i] = 0.0f;
}

// ---------------------------------------------------------------------------
// Scatter-add of h[src[e]] into accum[dst[e]] (+ optional degree count).
// One wave per edge (grid-stride): 2 coalesced float4 gathers + 8 f32 atomics
// per lane -> global_atomic_add_f32 at L2 (node table is L2-resident).
// ---------------------------------------------------------------------------
__global__ __launch_bounds__(256) void scatter_kernel(
        const float* __restrict__ h,
        const int* __restrict__ src,
        const int* __restrict__ dst,
        float* __restrict__ accum,
        float* __restrict__ deg,   // null = skip degree
        int n_edges) {
    const int lane   = threadIdx.x & 31;
    const int gwave  = (int)(((long)blockIdx.x * blockDim.x + threadIdx.x) >> 5);
    const int nwaves = (int)(((long)gridDim.x * blockDim.x) >> 5);
    for (int e = gwave; e < n_edges; e += nwaves) {
        int s = src[e];
        int d = dst[e];
        const float4* hs = (const float4*)(h + (long)s * FEATS);
        float* ac = accum + (long)d * FEATS;
        float4 x = hs[lane];
        atomicAdd(ac + 4 * lane + 0, x.x);
        atomicAdd(ac + 4 * lane + 1, x.y);
        atomicAdd(ac + 4 * lane + 2, x.z);
        atomicAdd(ac + 4 * lane + 3, x.w);
        float4 y = hs[lane + 32];
        atomicAdd(ac + 4 * (lane + 32) + 0, y.x);
        atomicAdd(ac + 4 * (lane + 32) + 1, y.y);
        atomicAdd(ac + 4 * (lane + 32) + 2, y.z);
        atomicAdd(ac + 4 * (lane + 32) + 3, y.w);
        if (deg != nullptr && lane == 0) atomicAdd(&deg[d], 1.0f);
    }
}

// ---------------------------------------------------------------------------
// accum[n][:] /= max(deg[n], 1), float4-vectorized (64 float4 per node row)
// ---------------------------------------------------------------------------
__global__ void normalize_kernel(float* __restrict__ accum,
                                 const float* __restrict__ deg,
                                 int total4) {
    int i = blockIdx.x * blockDim.x + threadIdx.x;
    if (i >= total4) return;
    int n = i >> 6;
    float inv = 1.0f / fmaxf(deg[n], 1.0f);
    float4* p = (float4*)accum;
    float4 v = p[i];
    v.x *= inv; v.y *= inv; v.z *= inv; v.w *= inv;
    p[i] = v;
}

// ---------------------------------------------------------------------------
// Fused dual-GEMM: Out = act(Hs*Ws + Hn*Wn + bias)
//   Hs, Hn : [M, 256] row-major    Ws, Wn : [256, 256] row-major (K x N)
// Block = 256 threads = 8 waves. Each block owns one 16-row M-strip and 8 of
// the 16 column tiles; both A tiles (16x256 Hs + Hn) are staged once in LDS
// (padded stride 260 -> all 32 lanes conflict-free on column reads) and
// shared by the 8 waves. Each wave runs the K-loop with two chained
// V_WMMA_F32_16X16X4_F32 (self + neighbor) per K-step of 4.
//
// Fragment layouts per CDNA5 ISA 7.12.2 (wave32):
//   A 16x4 f32: lane l -> row (l&15); VGPR0 = K[(l>>4)*2], VGPR1 = +1
//   B 4x16 f32: lane l -> col (l&15); same K striping
//   C/D 16x16 f32: VGPR v -> row v + (l>>4)*8, col (l&15)
// ---------------------------------------------------------------------------
template <bool RELU>
__global__ __launch_bounds__(256) void sage_gemm_kernel(
        const float* __restrict__ Hs,
        const float* __restrict__ Hn,
        const float* __restrict__ Ws,
        const float* __restrict__ Wn,
        const float* __restrict__ bias,
        float* __restrict__ Out,
        int M) {
    const int K = FEATS, N = FEATS;
    __shared__ float As[16 * LDA];
    __shared__ float An[16 * LDA];

    const int mt   = (int)blockIdx.x >> 1;       // M-tile
    const int half = (int)blockIdx.x & 1;        // which 8 column tiles
    const int m0   = mt * 16;
    const int wv   = threadIdx.x >> 5;
    const int lane = threadIdx.x & 31;
    const int n0   = (half * 8 + wv) * 16;
    if (m0 >= M) return;                         // uniform per block

    // Cooperative stage of both A tiles: 16 rows x 256 f32 = 1024 float4 each
    {
        const float4* gs = (const float4*)(Hs + (long)m0 * K);
        const float4* gn = (const float4*)(Hn + (long)m0 * K);
        for (int i = threadIdx.x; i < 1024; i += 256) {
            int r = i >> 6;                      // row (64 float4 per row)
            int c = i & 63;                      // float4 column
            *(float4*)&As[r * LDA + c * 4] = gs[i];
            *(float4*)&An[r * LDA + c * 4] = gn[i];
        }
    }
    __syncthreads();

    const int row  = lane & 15;
    const int col  = lane & 15;
    const int koff = (lane >> 4) << 1;           // 0 or 2

    v8f acc;
    float bv = bias[n0 + col];
#pragma unroll
    for (int v = 0; v < 8; ++v) acc[v] = bv;

    const float* as = As + row * LDA;
    const float* an = An + row * LDA;
    const float* ws = Ws + n0 + col;
    const float* wn = Wn + n0 + col;

#pragma unroll 8
    for (int k = 0; k < K; k += 4) {
        v2f a_s = { as[k + koff], as[k + koff + 1] };
        v2f a_n = { an[k + koff], an[k + koff + 1] };
        v2f b_s = { ws[(long)(k + koff) * N], ws[(long)(k + koff + 1) * N] };
        v2f b_n = { wn[(long)(k + koff) * N], wn[(long)(k + koff + 1) * N] };
        acc = __builtin_amdgcn_wmma_f32_16x16x4_f32(
                false, a_s, false, b_s, (short)0, acc, false, false);
        acc = __builtin_amdgcn_wmma_f32_16x16x4_f32(
                false, a_n, false, b_n, (short)0, acc, false, false);
    }

    const int rbase = m0 + ((lane >> 4) << 3);   // +8 for upper half-wave
#pragma unroll
    for (int v = 0; v < 8; ++v) {
        float x = acc[v];
        if (RELU) x = fmaxf(x, 0.0f);
        Out[(long)(rbase + v) * N + n0 + col] = x;
    }
}

// ---------------------------------------------------------------------------
// Per-edge dot product: out[e] = dot(h[src[e]], h[dst[e]])  (d = 256)
// One wave per edge; each lane covers 8 floats via two float4 loads.
// ---------------------------------------------------------------------------
__global__ __launch_bounds__(256) void edge_dot_kernel(
        const float* __restrict__ h,
        const int* __restrict__ src,
        const int* __restrict__ dst,
        float* __restrict__ out,
        int n_edges) {
    int wave = (int)(((long)blockIdx.x * blockDim.x + threadIdx.x) >> 5);
    int lane = threadIdx.x & 31;
    if (wave >= n_edges) return;
    int s = src[wave];
    int d = dst[wave];
    const float4* a = (const float4*)(h + (long)s * FEATS);
    const float4* b = (const float4*)(h + (long)d * FEATS);
    float4 x = a[lane];
    float4 y = b[lane];
    float acc = x.x * y.x + x.y * y.y + x.z * y.z + x.w * y.w;
    x = a[lane + 32];
    y = b[lane + 32];
    acc += x.x * y.x + x.y * y.y + x.z * y.z + x.w * y.w;
#pragma unroll
    for (int off = 16; off > 0; off >>= 1)
        acc += __shfl_down(acc, off, 32);
    if (lane == 0) out[wave] = acc;
}

// ---------------------------------------------------------------------------
// Launch
// ---------------------------------------------------------------------------
extern "C" void kernel_launch(void* const* d_in, const int* in_sizes, int n_in,
                              void* d_out, int out_size, void* d_ws, size_t ws_size,
                              hipStream_t stream) {
    const float* features = (const float*)d_in[0];
    const int*   src      = (const int*)d_in[1];
    const int*   dst      = (const int*)d_in[2];
    const float* W_self1  = (const float*)d_in[3];
    const float* W_neigh1 = (const float*)d_in[4];
    const float* b1       = (const float*)d_in[5];
    const float* W_self2  = (const float*)d_in[6];
    const float* W_neigh2 = (const float*)d_in[7];
    const float* b2       = (const float*)d_in[8];
    float* score = (float*)d_out;

    // Workspace layout (floats)
    const long NNF = (long)N_NODES * FEATS;         // 5,120,000
    float* neigh = (float*)d_ws;                    // [N_NODES, 256]
    float* deg   = neigh + NNF;                     // [N_NODES]
    float* h1    = deg + N_NODES;                   // [N_NODES, 256]
    float* h2    = h1 + NNF;                        // [N_NODES, 256]

    const int ZB = 256;
    const long z1 = NNF + N_NODES;                  // neigh + deg
    const int  scatter_blocks = 1024;               // 8192 grid-striding waves
    const int  norm_blocks    = (int)((NNF / 4 + ZB - 1) / ZB);
    const int  gemm_blocks    = (N_NODES / 16) * 2; // 2500
    const int  dot_blocks     = (int)(((long)N_EDGES * 32 + 255) / 256);

    // ---- Layer 1 ----
    zero_kernel<<<(int)((z1 + ZB - 1) / ZB), ZB, 0, stream>>>(neigh, z1);
    scatter_kernel<<<scatter_blocks, 256, 0, stream>>>(features, src, dst, neigh, deg, N_EDGES);
    normalize_kernel<<<norm_blocks, ZB, 0, stream>>>(neigh, deg, (int)(NNF / 4));
    sage_gemm_kernel<true><<<gemm_blocks, 256, 0, stream>>>(
            features, neigh, W_self1, W_neigh1, b1, h1, N_NODES);

    // ---- Layer 2 (degrees unchanged -> reuse deg) ----
    zero_kernel<<<(int)((NNF + ZB - 1) / ZB), ZB, 0, stream>>>(neigh, NNF);
    scatter_kernel<<<scatter_blocks, 256, 0, stream>>>(h1, src, dst, neigh, nullptr, N_EDGES);
    normalize_kernel<<<norm_blocks, ZB, 0, stream>>>(neigh, deg, (int)(NNF / 4));
    sage_gemm_kernel<false><<<gemm_blocks, 256, 0, stream>>>(
            h1, neigh, W_self2, W_neigh2, b2, h2, N_NODES);

    // ---- Edge scores ----
    edge_dot_kernel<<<dot_blocks, 256, 0, stream>>>(h2, src, dst, score, N_EDGES);
}